// Spline_50800873177328
// MI455X (gfx1250) — compile-verified
//
#include <hip/hip_runtime.h>

typedef float v2f __attribute__((ext_vector_type(2)));
typedef float v8f __attribute__((ext_vector_type(8)));

#define N_CTRL 64
#define N_KNOTS 68

// Cubic B-spline evaluation, WMMA-reduced.
// Each wave handles 32 contiguous points per iteration:
//   - every lane: load x, locate knot span s, local Cox-de Boor -> 4 weights
//   - two rounds (points 0-15 / 16-31): build sparse Basis(16x64) A-fragments
//     branch-free (each fragment element is nonzero in at most one K-slice),
//     accumulate out16 = Basis x c with 16 chained v_wmma_f32_16x16x4_f32,
//     store out[0..15] from lanes 0/16.
__global__ __launch_bounds__(256) void Spline_wmma_kernel(
    const float* __restrict__ x,
    const float* __restrict__ knots,
    const float* __restrict__ ctrl,
    float* __restrict__ out,
    int n)
{
    __shared__ float sc[N_CTRL];
    __shared__ float sk[N_KNOTS];

    const int tid  = threadIdx.x;
    const int lane = tid & 31;

    if (tid < N_CTRL)  sc[tid] = ctrl[tid];
    if (tid < N_KNOTS) sk[tid] = knots[tid];
    __syncthreads();

    const bool hi  = lane >= 16;          // lanes 16-31 carry K=2,3 / M=8..15
    const int  off = hi ? 2 : 0;

    // Loop-invariant B fragments: B[k'][ncol] = c[4k + k'] for every column.
    // f32 4x16 B layout: v0 = {K0 | K2}, v1 = {K1 | K3} across lane halves.
    v2f bfrag[16];
#pragma unroll
    for (int k = 0; k < 16; ++k) {
        bfrag[k].x = sc[4 * k + off];
        bfrag[k].y = sc[4 * k + off + 1];
    }

    const int stride = (int)(gridDim.x * blockDim.x);
    int pt = (int)(blockIdx.x * blockDim.x) + tid;

    for (int waveBase = pt - lane; waveBase < n;
         waveBase += stride, pt += stride) {

        // Clamped tail load keeps every lane active (EXEC all-ones for WMMA).
        const int li = (pt < n) ? pt : (n - 1);
        const float xv = x[li];

        // ---- knot span search: t[s] <= x < t[s+1], s in [0,66] ----
        int s = (int)(xv * 67.0f);
        s = s < 0 ? 0 : (s > 66 ? 66 : s);
        while (s > 0 && xv < sk[s]) --s;
        while (s < 66 && xv >= sk[s + 1]) ++s;

        // ---- local Cox-de Boor: Nb[r] = B_{s-3+r,3}(x) ----
        // Knot indices clamped: out-of-range reads only pollute entries whose
        // columns (<0 or >63) never materialize in the A fragments.
        float Nb[4];
        Nb[0] = 1.0f; Nb[1] = 0.0f; Nb[2] = 0.0f; Nb[3] = 0.0f;
        float left[4], right[4];
#pragma unroll
        for (int j = 1; j <= 3; ++j) {
            int il = s + 1 - j; il = il < 0 ? 0 : il;
            int ir = s + j;     ir = ir > (N_KNOTS - 1) ? (N_KNOTS - 1) : ir;
            left[j]  = xv - sk[il];
            right[j] = sk[ir] - xv;
            float saved = 0.0f;
#pragma unroll
            for (int r = 0; r < j; ++r) {
                float temp = Nb[r] / (right[r + 1] + left[j - r]);
                Nb[r] = saved + right[r + 1] * temp;
                saved = left[j - r] * temp;
            }
            Nb[j] = saved;
        }

        // ---- two WMMA rounds: source points = lanes 0-15, then 16-31 ----
#pragma unroll
        for (int rb = 0; rb < 2; ++rb) {
            const int srcBase = rb * 16;
            const int src = (lane & 15) + srcBase;

            const int   ss = __shfl(s,     src, 32);
            const float u0 = __shfl(Nb[0], src, 32);
            const float u1 = __shfl(Nb[1], src, 32);
            const float u2 = __shfl(Nb[2], src, 32);
            const float u3 = __shfl(Nb[3], src, 32);

            // Column col = 4k + off holds weight index j0 = 4k + jb.
            // j0 lands in [0,3] for exactly one k (j0 steps by 4):
            //   jx = j0 at the hit, kx = the hit slice (out of [0,15] if
            //   truncated at the spline ends -> element stays zero).
            const int jb = off + 3 - ss;
            const int jx = jb & 3;
            const int kx = (jx - jb) >> 2;
            const int jy = (jb + 1) & 3;
            const int ky = (jy - (jb + 1)) >> 2;
            const float vx = (jx == 0) ? u0 : (jx == 1) ? u1
                           : (jx == 2) ? u2 : u3;
            const float vy = (jy == 0) ? u0 : (jy == 1) ? u1
                           : (jy == 2) ? u2 : u3;

            v8f acc = {0.f, 0.f, 0.f, 0.f, 0.f, 0.f, 0.f, 0.f};
#pragma unroll
            for (int k = 0; k < 16; ++k) {
                v2f afrag;
                afrag.x = (k == kx) ? vx : 0.0f;   // single v_cndmask each
                afrag.y = (k == ky) ? vy : 0.0f;
                acc = __builtin_amdgcn_wmma_f32_16x16x4_f32(
                    /*neg_a=*/false, afrag, /*neg_b=*/false, bfrag[k],
                    /*c_mod=*/(short)0, acc,
                    /*reuse_a=*/false, /*reuse_b=*/false);
            }

            // All D columns are identical (B columns identical):
            // lane 0 : v0..7 = out[0..7], lane 16 : v0..7 = out[8..15].
            const int gb = waveBase + srcBase + (hi ? 8 : 0);
            if ((lane & 15) == 0) {
                if (gb + 8 <= n) {
                    float4 lo4 = { acc[0], acc[1], acc[2], acc[3] };
                    float4 hi4 = { acc[4], acc[5], acc[6], acc[7] };
                    *(float4*)(out + gb)     = lo4;
                    *(float4*)(out + gb + 4) = hi4;
                } else {
#pragma unroll
                    for (int m = 0; m < 8; ++m) {
                        if (gb + m < n) out[gb + m] = acc[m];
                    }
                }
            }
        }
    }
}

extern "C" void kernel_launch(void* const* d_in, const int* in_sizes, int n_in,
                              void* d_out, int out_size, void* d_ws, size_t ws_size,
                              hipStream_t stream) {
    const float* x     = (const float*)d_in[0];
    const float* knots = (const float*)d_in[1];
    const float* ctrl  = (const float*)d_in[2];
    float*       out   = (float*)d_out;
    const int n = in_sizes[0];

    const int threads = 256;
    int blocks = (n + threads - 1) / threads;
    if (blocks < 1) blocks = 1;
    if (blocks > 16384) blocks = 16384;   // grid-stride covers the rest

    hipLaunchKernelGGL(Spline_wmma_kernel, dim3(blocks), dim3(threads), 0, stream,
                       x, knots, ctrl, out, n);
}